// DistGraphConv_51032801411439
// MI455X (gfx1250) — compile-verified
//
#include <hip/hip_runtime.h>

// GCN layer for MI455X (gfx1250, wave32).
// Phase breakdown:
//   K1 zero workspace
//   K2 degree count (f32 atomics)
//   K3 deg -> rsqrt(max(deg,1)) in place
//   K4 edge gather/scale/scatter-add (wave-per-edge, float4 loads, f32 atomics)
//   K5 agg @ W via v_wmma_f32_16x16x4_f32 + right-norm + bias + relu

#define IN_FEATS  128
#define OUT_FEATS 128

typedef float v2f __attribute__((ext_vector_type(2)));
typedef float v8f __attribute__((ext_vector_type(8)));

__global__ void k_zero(float* __restrict__ p, int n) {
    int i = blockIdx.x * blockDim.x + threadIdx.x;
    if (i < n) p[i] = 0.0f;
}

__global__ void k_degrees(const int* __restrict__ src, const int* __restrict__ dst,
                          float* __restrict__ out_deg, float* __restrict__ in_deg, int E) {
    int i = blockIdx.x * blockDim.x + threadIdx.x;
    if (i < E) {
        atomicAdd(&out_deg[src[i]], 1.0f);
        atomicAdd(&in_deg[dst[i]], 1.0f);
    }
}

__global__ void k_invsqrt(float* __restrict__ p, int n) {
    int i = blockIdx.x * blockDim.x + threadIdx.x;
    if (i < n) p[i] = rsqrtf(fmaxf(p[i], 1.0f));
}

// One wave (32 lanes) per edge; lane covers 4 consecutive features.
__global__ __launch_bounds__(256)
void k_scatter(const float* __restrict__ feat, const int* __restrict__ src,
               const int* __restrict__ dst, const float* __restrict__ out_scale,
               float* __restrict__ agg, int E) {
    int e = blockIdx.x * 8 + (threadIdx.x >> 5);
    if (e >= E) return;
    int lane = threadIdx.x & 31;
    int s = src[e];
    int d = dst[e];
    float sc = out_scale[s];
    float4 v = ((const float4*)(feat + (size_t)s * IN_FEATS))[lane];
    float* ap = agg + (size_t)d * IN_FEATS + lane * 4;
    atomicAdd(ap + 0, v.x * sc);
    atomicAdd(ap + 1, v.y * sc);
    atomicAdd(ap + 2, v.z * sc);
    atomicAdd(ap + 3, v.w * sc);
}

// One wave per 16x16 output tile. Block = 256 threads = 8 waves = one row tile
// (16 nodes) x all 8 column tiles (128 out feats).
// A-fragment (16x4 f32): lanes 0-15 -> M=lane, K = k+0 / k+1 in vgpr0/1;
// lanes 16-31 -> M=lane-16, K = k+2 / k+3. B mirrors with N=lane&15.
// C/D: vgpr v -> M = v + 8*(lane>=16), N = lane&15.
__global__ __launch_bounds__(256)
void k_gemm_wmma(const float* __restrict__ agg, const float* __restrict__ W,
                 const float* __restrict__ in_scale, const float* __restrict__ bias,
                 float* __restrict__ out) {
    int lane = threadIdx.x & 31;
    int row0 = blockIdx.x * 16;           // row tile (16 nodes)
    int col0 = (threadIdx.x >> 5) * 16;   // wave id in block -> column tile
    int m  = lane & 15;                   // M index for A, N index for B
    int hi = lane >> 4;                   // 0: K pair (0,1); 1: K pair (2,3)

    const float* arow = agg + (size_t)(row0 + m) * IN_FEATS + hi * 2;

    v8f c = {};
#pragma unroll
    for (int k = 0; k < IN_FEATS; k += 4) {
        v2f a;
        a.x = arow[k];
        a.y = arow[k + 1];
        v2f b;
        b.x = W[(size_t)(k + hi * 2 + 0) * OUT_FEATS + col0 + m];
        b.y = W[(size_t)(k + hi * 2 + 1) * OUT_FEATS + col0 + m];
        c = __builtin_amdgcn_wmma_f32_16x16x4_f32(
            /*neg_a=*/false, a, /*neg_b=*/false, b,
            /*c_mod=*/(short)0, c, /*reuse_a=*/false, /*reuse_b=*/false);
    }

    float bv = bias[col0 + m];
#pragma unroll
    for (int v = 0; v < 8; ++v) {
        int row = row0 + v + hi * 8;
        float r = c[v] * in_scale[row] + bv;
        out[(size_t)row * OUT_FEATS + col0 + m] = fmaxf(r, 0.0f);
    }
}

extern "C" void kernel_launch(void* const* d_in, const int* in_sizes, int n_in,
                              void* d_out, int out_size, void* d_ws, size_t ws_size,
                              hipStream_t stream) {
    const float* feat   = (const float*)d_in[0];
    const int*   src    = (const int*)d_in[1];
    const int*   dst    = (const int*)d_in[2];
    const float* weight = (const float*)d_in[3];
    const float* bias   = (const float*)d_in[4];
    float*       out    = (float*)d_out;

    const int N = in_sizes[0] / IN_FEATS;   // 50000
    const int E = in_sizes[1];              // 600000

    float* agg     = (float*)d_ws;               // N * 128
    float* out_deg = agg + (size_t)N * IN_FEATS; // N
    float* in_deg  = out_deg + N;                // N (contiguous with out_deg)

    // 1) zero agg + both degree arrays
    int nz = N * IN_FEATS + 2 * N;
    k_zero<<<(nz + 255) / 256, 256, 0, stream>>>(agg, nz);

    // 2) degrees
    k_degrees<<<(E + 255) / 256, 256, 0, stream>>>(src, dst, out_deg, in_deg, E);

    // 3) deg -> deg^-0.5 (clamped); out_deg and in_deg are contiguous
    k_invsqrt<<<(2 * N + 255) / 256, 256, 0, stream>>>(out_deg, 2 * N);

    // 4) gather + left-norm + scatter-add (wave per edge, 8 edges per block)
    k_scatter<<<(E + 7) / 8, 256, 0, stream>>>(feat, src, dst, out_deg, agg, E);

    // 5) WMMA GEMM + right-norm + bias + relu. N is a multiple of 16 (50000 = 3125*16).
    k_gemm_wmma<<<N / 16, 256, 0, stream>>>(agg, weight, in_deg, bias, out);
}